// QuantizationLayer_21345987461308
// MI455X (gfx1250) — compile-verified
//
#include <hip/hip_runtime.h>
#include <hip/hip_bf16.h>

typedef __attribute__((ext_vector_type(2))) float v2f;
typedef __attribute__((ext_vector_type(8))) float v8f;

#define N_TOK   65536
#define DIM     256
#define KCODE   1024
#define HWSZ    1024           // H*W
#define BPLANE  262144         // D*H*W floats per batch image
#define OUT_HALF 16777216      // B*D*H*W

// ---------------------------------------------------------------------------
// Kernel 0: codebook squared norms  csq[k] = sum_d cb[k][d]^2
// 1024 blocks x 32 threads (one wave per code)
// ---------------------------------------------------------------------------
__global__ __launch_bounds__(32) void vq_csq(const float* __restrict__ cb,
                                             float* __restrict__ csq) {
    int k = blockIdx.x;
    int l = threadIdx.x;                      // 0..31
    const float4* row = (const float4*)(cb + (size_t)k * DIM);
    float4 a = row[l];
    float4 b = row[l + 32];
    float s = a.x * a.x + a.y * a.y + a.z * a.z + a.w * a.w
            + b.x * b.x + b.y * b.y + b.z * b.z + b.w * b.w;
#pragma unroll
    for (int off = 16; off >= 1; off >>= 1)
        s += __shfl_xor(s, off, 32);
    if (l == 0) csq[k] = s;
}

// ---------------------------------------------------------------------------
// Kernel 1: fused GEMM + argmin.
//   score[n][k] = ||c_k||^2 - 2 * (x_n . c_k)      (||x_n||^2 dropped: const per n)
// Block = 256 threads = 8 waves; each wave owns a 16-token tile and computes
// TWO 16x16 output tiles per chunk (32 codes staged in LDS) with two
// independent v_wmma_f32_16x16x4_f32 accumulator chains sharing the A regs.
// ---------------------------------------------------------------------------
__global__ __launch_bounds__(256) void vq_argmin(const float* __restrict__ x,
                                                 const float* __restrict__ cb,
                                                 const float* __restrict__ csq,
                                                 int* __restrict__ idxbuf) {
    __shared__ float ldsB[32 * 260];   // 32 codes x 256 d, rows padded to 260
    __shared__ float ldsCsq[KCODE];

    const int tid  = threadIdx.x;
    const int lane = tid & 31;
    const int lm   = lane & 15;        // column / row-in-halfwave selector
    const int half = lane >> 4;        // 0: K=0,1 ; 1: K=2,3 of each 4-chunk
    const int wv   = tid >> 5;         // wave id 0..7
    const int dsel = half * 2;

    // stage codebook norms once
    for (int i = tid; i < KCODE; i += 256) ldsCsq[i] = csq[i];

    // ---- load A fragments for this wave's 16 tokens, all of D, into regs ----
    const int t0  = blockIdx.x * 128 + wv * 16;   // token tile base (16-aligned)
    const int b   = t0 >> 10;
    const int hw0 = t0 & (HWSZ - 1);
    // element (m = lm, d = 4c + dsel + {0,1}) at  x[b*BPLANE + d*HWSZ + hw0 + m]
    const float* pax = x + (size_t)b * BPLANE + (size_t)dsel * HWSZ + hw0 + lm;

    v2f areg[64];
#pragma unroll
    for (int c = 0; c < 64; ++c) {
        areg[c].x = pax[c * 4096];          // d = 4c + dsel
        areg[c].y = pax[c * 4096 + 1024];   // d = 4c + dsel + 1
    }

    float minv[8];
    int   mini[8];
#pragma unroll
    for (int r = 0; r < 8; ++r) { minv[r] = 3.4e38f; mini[r] = 0; }

    const float* bp0 = &ldsB[lm * 260 + dsel];          // codes k0 + 0..15
    const float* bp1 = &ldsB[(16 + lm) * 260 + dsel];   // codes k0 + 16..31

    for (int chunk = 0; chunk < 32; ++chunk) {
        const int k0 = chunk << 5;
        __syncthreads();
        // stage 32 codebook rows (contiguous 32 KB) into LDS, padded rows
        {
            const float4* src = (const float4*)(cb + (size_t)k0 * DIM);
#pragma unroll
            for (int i = 0; i < 8; ++i) {
                int q   = i * 256 + tid;    // float4 index in [0,2048)
                int row = q >> 6;
                int c4  = q & 63;
                *(float4*)&ldsB[row * 260 + c4 * 4] = src[q];
            }
        }
        __syncthreads();

        v8f acc0 = {};
        v8f acc1 = {};
#pragma unroll
        for (int c = 0; c < 64; ++c) {
            v2f bf0 = *(const v2f*)(bp0 + c * 4);
            v2f bf1 = *(const v2f*)(bp1 + c * 4);
            acc0 = __builtin_amdgcn_wmma_f32_16x16x4_f32(
                false, areg[c], false, bf0, (short)0, acc0, false, false);
            acc1 = __builtin_amdgcn_wmma_f32_16x16x4_f32(
                false, areg[c], false, bf1, (short)0, acc1, false, false);
        }

        // epilogue: C/D layout -> VGPR r holds row M=r (lanes 0-15) / M=r+8
        // (lanes 16-31); this lane's columns are k0+lm and k0+16+lm.
        const int   kn0 = k0 + lm;
        const int   kn1 = k0 + 16 + lm;
        const float cv0 = ldsCsq[kn0];
        const float cv1 = ldsCsq[kn1];
#pragma unroll
        for (int r = 0; r < 8; ++r) {
            float s0 = cv0 - 2.0f * acc0[r];
            float s1 = cv1 - 2.0f * acc1[r];
            if (s0 < minv[r]) { minv[r] = s0; mini[r] = kn0; }  // ascending k:
            if (s1 < minv[r]) { minv[r] = s1; mini[r] = kn1; }  // first-min kept
        }
    }

    // cross-lane argmin over the 16 lanes of each half-wave
#pragma unroll
    for (int r = 0; r < 8; ++r) {
        float v  = minv[r];
        int   id = mini[r];
#pragma unroll
        for (int off = 8; off >= 1; off >>= 1) {
            float ov = __shfl_xor(v, off, 16);
            int   oi = __shfl_xor(id, off, 16);
            if (ov < v || (ov == v && oi < id)) { v = ov; id = oi; }
        }
        minv[r] = v; mini[r] = id;
    }

    if (lm == 0) {                       // lanes 0 (rows 0-7) and 16 (rows 8-15)
        int base = t0 + half * 8;
#pragma unroll
        for (int r = 0; r < 8; ++r) idxbuf[base + r] = mini[r];
    }
}

// ---------------------------------------------------------------------------
// Kernel 2: gather + straight-through output.
//   f = n*256 + d ;  q = cb[idx[n]][d]
//   out[f]            = x[f] + (q - x[f])   (z_hat, fp32-rounded like the ref)
//   out[OUT_HALF + f] = q                   (z_q)
// float4 vectorized: 4,194,304 float4 elements.
// ---------------------------------------------------------------------------
__global__ __launch_bounds__(256) void vq_out(const float* __restrict__ x,
                                              const float* __restrict__ cb,
                                              const int* __restrict__ idxbuf,
                                              float* __restrict__ out) {
    int i = blockIdx.x * 256 + threadIdx.x;   // float4 index
    int f = i << 2;
    int n = f >> 8;
    int d = f & 255;
    int k = idxbuf[n];
    float4 q  = *(const float4*)(cb + (size_t)k * DIM + d);
    float4 xv = *(const float4*)(x + f);
    float4 zh;
    zh.x = xv.x + (q.x - xv.x);
    zh.y = xv.y + (q.y - xv.y);
    zh.z = xv.z + (q.z - xv.z);
    zh.w = xv.w + (q.w - xv.w);
    *(float4*)(out + f)            = zh;
    *(float4*)(out + OUT_HALF + f) = q;
}

// ---------------------------------------------------------------------------
extern "C" void kernel_launch(void* const* d_in, const int* in_sizes, int n_in,
                              void* d_out, int out_size, void* d_ws, size_t ws_size,
                              hipStream_t stream) {
    const float* x  = (const float*)d_in[0];   // [64,256,32,32]
    const float* cb = (const float*)d_in[1];   // [1024,256]
    float* out = (float*)d_out;                // [2 * 16777216]

    float* csq   = (float*)d_ws;                        // 1024 floats
    int*   idxbf = (int*)((char*)d_ws + 4096);          // 65536 ints

    vq_csq<<<KCODE, 32, 0, stream>>>(cb, csq);
    vq_argmin<<<N_TOK / 128, 256, 0, stream>>>(x, cb, csq, idxbf);
    vq_out<<<(N_TOK * DIM / 4) / 256, 256, 0, stream>>>(x, cb, idxbf, out);
}